// KbNufft_46093589021406
// MI455X (gfx1250) — compile-verified
//
#include <hip/hip_runtime.h>
#include <math.h>

typedef __attribute__((ext_vector_type(2))) float v2f;
typedef __attribute__((ext_vector_type(8))) float v8f;

#define GRID_K 512
#define NB 2
#define NC 16
#define NM 131072
#define NPLANES (NB*NC)          // 32 images
#define PLANE (GRID_K*GRID_K)    // 262144
#define TABC 3072                // J*L/2
#define TABMAX 6144              // J*L

// ---------------------------------------------------------------------------
// WMMA helpers: 16x16x16 complex matmul built from V_WMMA_F32_16X16X4_F32
// ---------------------------------------------------------------------------
static __device__ __forceinline__ v8f wmma4(v2f a, v2f b, v8f c) {
  return __builtin_amdgcn_wmma_f32_16x16x4_f32(false, a, false, b, (short)0, c,
                                               false, false);
}

// (Cr,Ci) += A(16x16 cplx) * B(16x16 cplx); A row-major [m][k] (ld=lda),
// B row-major [k][n] (ld=ldb), both in LDS. K=16 as four chained k=4 WMMAs.
// A frag layout (ISA 7.12.2): lanes 0-15 hold K={4kk,4kk+1}, lanes 16-31 K={4kk+2,4kk+3}.
static __device__ __forceinline__ void cwmma16(
    const float* Ar, const float* Ai, int lda,
    const float* Br, const float* Bi, int ldb,
    v8f& Cr, v8f& Ci, int lane)
{
  const int ln = lane & 15;
  const int hi = (lane >> 4) & 1;
#pragma unroll
  for (int kk = 0; kk < 4; ++kk) {
    const int k = kk * 4 + hi * 2;
    v2f ar, ai, br, bi;
    ar.x = Ar[ln * lda + k];      ar.y = Ar[ln * lda + k + 1];
    ai.x = Ai[ln * lda + k];      ai.y = Ai[ln * lda + k + 1];
    br.x = Br[k * ldb + ln];      br.y = Br[(k + 1) * ldb + ln];
    bi.x = Bi[k * ldb + ln];      bi.y = Bi[(k + 1) * ldb + ln];
    v2f nai = -ai;                 // f32 WMMA NEG only allows CNeg -> negate in VALU
    Cr = wmma4(ar, br, Cr);
    Cr = wmma4(nai, bi, Cr);
    Ci = wmma4(ar, bi, Ci);
    Ci = wmma4(ai, br, Ci);
  }
}

// Store 16x16 f32 C/D frag (VGPR r: lanes<16 -> M=r, lanes>=16 -> M=r+8)
static __device__ __forceinline__ void storeC(float* Or, float* Oi, int rs, int cs,
                                              v8f cr, v8f ci, int lane)
{
  const int cn = lane & 15;
  const int hi = (lane >> 4) & 1;
#pragma unroll
  for (int r = 0; r < 8; ++r) {
    const int row = r + hi * 8;
    Or[row * rs + cn * cs] = cr[r];
    Oi[row * rs + cn * cs] = ci[r];
  }
}

// ---------------------------------------------------------------------------
// Kernel 1: apodize (x * scaling_coef) and zero-pad into SoA 512x512 planes
// ---------------------------------------------------------------------------
__global__ void apod_pad(const float* __restrict__ x, const float* __restrict__ sc,
                         float* __restrict__ gRe, float* __restrict__ gIm)
{
  long tid = (long)blockIdx.x * blockDim.x + threadIdx.x;  // NPLANES*PLANE threads
  int p   = (int)(tid >> 18);
  int rem = (int)(tid & (PLANE - 1));
  int y = rem >> 9, xc = rem & 511;
  float gr = 0.f, gi = 0.f;
  if (y < 256 && xc < 256) {
    long ib = ((long)p * 2) * 65536 + (long)y * 256 + xc;   // (b,c,re/im,256,256)
    float xre = x[ib], xim = x[ib + 65536];
    int si = (y * 256 + xc) * 2;                             // complex64 interleaved
    float sr = sc[si], sii = sc[si + 1];
    gr = xre * sr - xim * sii;
    gi = xre * sii + xim * sr;
  }
  gRe[tid] = gr;
  gIm[tid] = gi;
}

// ---------------------------------------------------------------------------
// Kernel 2/3: 512-pt complex FFT per wave, Cooley-Tukey 32x16, WMMA stages.
// isCol=0: FFT along rows (only 256 nonzero rows/image); isCol=1: columns.
// ---------------------------------------------------------------------------
__launch_bounds__(128)
__global__ void fft512_lines(float* gRe, float* gIm, int isCol)
{
  __shared__ float sW32r[1024], sW32i[1024];   // DFT-32 matrix
  __shared__ float sW16r[256],  sW16i[256];    // DFT-16 matrix
  __shared__ float sTwr[512],   sTwi[512];     // twiddles w512^(n2*k1)
  __shared__ float sXr[4][512], sXi[4][512];   // per-wave in/out buffer
  __shared__ float sYr[4][512], sYi[4][512];   // per-wave stage buffer

  const int tid = threadIdx.x;
  const float TWO_PI = 6.28318530717958647692f;
  for (int t = tid; t < 1024; t += 128) {
    int a = t >> 5, b = t & 31;
    float s, c; __sincosf(-TWO_PI * (float)((a * b) & 31) * (1.f / 32.f), &s, &c);
    sW32r[t] = c; sW32i[t] = s;
  }
  for (int t = tid; t < 256; t += 128) {
    int a = t >> 4, b = t & 15;
    float s, c; __sincosf(-TWO_PI * (float)((a * b) & 15) * (1.f / 16.f), &s, &c);
    sW16r[t] = c; sW16i[t] = s;
  }
  for (int t = tid; t < 512; t += 128) {
    int k1 = t >> 4, n2 = t & 15;
    float s, c; __sincosf(-TWO_PI * (float)((k1 * n2) & 511) * (1.f / 512.f), &s, &c);
    sTwr[t] = c; sTwi[t] = s;
  }
  __syncthreads();

  const int w = tid >> 5, lane = tid & 31;
  const int line = blockIdx.x * 4 + w;
  long base; int stride;
  if (isCol) { int img = line >> 9; int col = line & 511;
               base = (long)img * PLANE + col;            stride = GRID_K; }
  else       { int img = line >> 8; int row = line & 255;
               base = (long)img * PLANE + (long)row * GRID_K; stride = 1; }

  float* xr = sXr[w]; float* xi = sXi[w];
  float* yr = sYr[w]; float* yi = sYi[w];
  for (int e = lane; e < 512; e += 32) {
    xr[e] = gRe[base + (long)e * stride];
    xi[e] = gIm[base + (long)e * stride];
  }
  __syncthreads();

  // Stage 1: Y(32x16) = W32 * Xhat,  Xhat[n1][n2] = x[16*n1 + n2]
#pragma unroll
  for (int i = 0; i < 2; ++i) {
    v8f cr = {}; v8f ci = {};
#pragma unroll
    for (int jb = 0; jb < 2; ++jb) {
      cwmma16(&sW32r[(i * 16) * 32 + jb * 16], &sW32i[(i * 16) * 32 + jb * 16], 32,
              &xr[jb * 256], &xi[jb * 256], 16, cr, ci, lane);
    }
    storeC(&yr[i * 256], &yi[i * 256], 16, 1, cr, ci, lane);   // Y[k1][n2]
  }
  __syncthreads();

  // Twiddle: T[k1][n2] = Y[k1][n2] * exp(-2i*pi*k1*n2/512)
  for (int t = lane; t < 512; t += 32) {
    float a = yr[t], b = yi[t];
    float trn = sTwr[t], tin = sTwi[t];
    yr[t] = a * trn - b * tin;
    yi[t] = a * tin + b * trn;
  }
  __syncthreads();

  // Stage 2: Z(32x16) = T * W16 ; output X[k1 + 32*k2] = Z[k1][k2]
#pragma unroll
  for (int i = 0; i < 2; ++i) {
    v8f cr = {}; v8f ci = {};
    cwmma16(&yr[i * 256], &yi[i * 256], 16, sW16r, sW16i, 16, cr, ci, lane);
    storeC(&xr[i * 16], &xi[i * 16], 1, 32, cr, ci, lane);
  }
  __syncthreads();

  for (int e = lane; e < 512; e += 32) {
    gRe[base + (long)e * stride] = xr[e];
    gIm[base + (long)e * stride] = xi[e];
  }
}

// ---------------------------------------------------------------------------
// Kernel 4: 6x6 KB interpolation for all 16 coils + fftshift phase
// ---------------------------------------------------------------------------
__launch_bounds__(256)
__global__ void interp6x6(const float* __restrict__ om,
                          const float* __restrict__ t0,
                          const float* __restrict__ t1,
                          const float* __restrict__ gRe,
                          const float* __restrict__ gIm,
                          float* __restrict__ out)
{
  int tid = blockIdx.x * 256 + threadIdx.x;        // NB*NM threads
  int b = tid >> 17;                               // NM = 2^17
  int m = tid & (NM - 1);
  float om0 = om[((long)b * 2 + 0) * NM + m];
  float om1 = om[((long)b * 2 + 1) * NM + m];
  const float K_OVER_2PI = 512.f / 6.28318530717958647692f;
  float tm0 = om0 * K_OVER_2PI, tm1 = om1 * K_OVER_2PI;
  int base0 = (int)floorf(tm0 - 3.f) + 1;
  int base1 = (int)floorf(tm1 - 3.f) + 1;

  float c0r[6], c0i[6]; int g0[6];
  float c1r[6], c1i[6]; int g1[6];
#pragma unroll
  for (int j = 0; j < 6; ++j) {
    int gi0 = base0 + j;
    int d0 = (int)rintf((tm0 - (float)gi0) * 1024.f) + TABC;
    d0 = min(max(d0, 0), TABMAX);
    c0r[j] = t0[2 * d0]; c0i[j] = t0[2 * d0 + 1];
    g0[j] = (gi0 & 511) * GRID_K;
    int gi1 = base1 + j;
    int d1 = (int)rintf((tm1 - (float)gi1) * 1024.f) + TABC;
    d1 = min(max(d1, 0), TABMAX);
    c1r[j] = t1[2 * d1]; c1i[j] = t1[2 * d1 + 1];
    g1[j] = gi1 & 511;
  }

  float ar[NC], ai[NC];
#pragma unroll
  for (int c = 0; c < NC; ++c) { ar[c] = 0.f; ai[c] = 0.f; }

  const long pbase = (long)b * NC * PLANE;
#pragma unroll
  for (int j0 = 0; j0 < 6; ++j0) {
#pragma unroll
    for (int j1 = 0; j1 < 6; ++j1) {
      float cr = c0r[j0] * c1r[j1] - c0i[j0] * c1i[j1];
      float ci = c0r[j0] * c1i[j1] + c0i[j0] * c1r[j1];
      long flat = pbase + g0[j0] + g1[j1];
#pragma unroll
      for (int c = 0; c < NC; ++c) {        // L2-resident gathers (grid = 64MB)
        float vr = gRe[flat + (long)c * PLANE];
        float vi = gIm[flat + (long)c * PLANE];
        ar[c] += cr * vr - ci * vi;
        ai[c] += cr * vi + ci * vr;
      }
    }
  }

  float ph = (om0 + om1) * 128.f;            // om . n_shift
  float sp, cp; __sincosf(ph, &sp, &cp);
#pragma unroll
  for (int c = 0; c < NC; ++c) {
    float yr2 = ar[c] * cp - ai[c] * sp;
    float yi2 = ar[c] * sp + ai[c] * cp;
    long ob = (((long)b * NC + c) * 2) * NM + m;
    out[ob] = yr2;
    out[ob + NM] = yi2;
  }
}

// ---------------------------------------------------------------------------
extern "C" void kernel_launch(void* const* d_in, const int* in_sizes, int n_in,
                              void* d_out, int out_size, void* d_ws, size_t ws_size,
                              hipStream_t stream) {
  (void)in_sizes; (void)n_in; (void)out_size; (void)ws_size;
  const float* x  = (const float*)d_in[0];   // (B,C,2,256,256) f32
  const float* om = (const float*)d_in[1];   // (B,2,M) f32
  const float* sc = (const float*)d_in[2];   // (256,256) complex64
  const float* t0 = (const float*)d_in[3];   // (6145,) complex64
  const float* t1 = (const float*)d_in[4];   // (6145,) complex64
  float* outp = (float*)d_out;

  float* gRe = (float*)d_ws;                         // 32 MB re plane
  float* gIm = gRe + (size_t)NPLANES * PLANE;        // 32 MB im plane

  apod_pad<<<NPLANES * PLANE / 256, 256, 0, stream>>>(x, sc, gRe, gIm);
  fft512_lines<<<(NPLANES * 256) / 4, 128, 0, stream>>>(gRe, gIm, 0); // rows
  fft512_lines<<<(NPLANES * 512) / 4, 128, 0, stream>>>(gRe, gIm, 1); // cols
  interp6x6<<<(NB * NM) / 256, 256, 0, stream>>>(om, t0, t1, gRe, gIm, outp);
}